// GCNBackbone_44805098832142
// MI455X (gfx1250) — compile-verified
//
#include <hip/hip_runtime.h>
#include <hip/hip_bf16.h>

#define N_NODES 50000
#define N_EDGES 800000
#define DIN 128
#define DOUT 128
#define LDS_STRIDE 132   // 128 + 4 pad: 16B-aligned rows, conflict-free ds_load_b64

typedef __attribute__((ext_vector_type(2))) float v2f;
typedef __attribute__((ext_vector_type(8))) float v8f;

// ---------------------------------------------------------------------------
// GEMM: C[nrows x 128] = A[nrows x 128] * W[128 x 128] via V_WMMA_F32_16X16X4_F32
// Block = 256 threads = 8 waves. Block tile = 32 rows x 128 cols.
// A tile staged in LDS once (cuts 8x redundant global A reads); each wave owns
// a 16-column strip and computes TWO 16x16 tiles (rows m0/m0+16) that share
// each B fragment -> 2 independent WMMA chains per wave.
// A frag: lane l -> A[(l&15)][k + (l>>4)*2 + j]        (ds_load_b64)
// B frag: lane l -> W[k + (l>>4)*2 + j][n0 + (l&15)]   (global b32 x2)
// C/D:    vgpr j, lane l -> C[m + j + 8*(l>>4)][n0 + (l&15)]
// ---------------------------------------------------------------------------
__launch_bounds__(256)
__global__ void gemm_wmma_f32(const float* __restrict__ A,
                              const float* __restrict__ W,
                              float* __restrict__ C, int nrows) {
  __shared__ float As[32 * LDS_STRIDE];   // 16,896 B

  const int tid  = threadIdx.x;
  const int lane = tid & 31;
  const int wave = tid >> 5;              // 0..7 -> N tile
  const int m0   = blockIdx.x * 32;
  const int rows = (nrows - m0) < 32 ? (nrows - m0) : 32;   // 32 or 16

  // Cooperative stage of the A tile: rows*32 float4 chunks across 256 threads.
  {
    const float4* src4 = (const float4*)(A + (size_t)m0 * DIN);
    const int n4 = rows * (DIN / 4);
    for (int i = tid; i < n4; i += 256) {
      const int r = i >> 5;
      const int c = i & 31;
      ((float4*)(As + r * LDS_STRIDE))[c] = src4[i];
    }
  }
  __syncthreads();

  const int n0    = wave * 16;
  const int mlane = lane & 15;
  const int khalf = (lane >> 4) * 2;      // 0 or 2

  const float* a0   = As + (size_t)mlane * LDS_STRIDE + khalf;        // rows 0..15
  const float* a1   = a0 + 16 * LDS_STRIDE;                           // rows 16..31
  const float* wcol = W + (size_t)khalf * DOUT + n0 + mlane;

  v8f acc0 = {}, acc1 = {};
#pragma unroll 4
  for (int k = 0; k < DIN; k += 4) {
    v2f b;
    b.x = wcol[(size_t)k * DOUT];
    b.y = wcol[(size_t)(k + 1) * DOUT];
    v2f x0 = *(const v2f*)(a0 + k);       // ds_load_b64
    v2f x1 = *(const v2f*)(a1 + k);
    acc0 = __builtin_amdgcn_wmma_f32_16x16x4_f32(false, x0, false, b,
                                                 (short)0, acc0, false, false);
    acc1 = __builtin_amdgcn_wmma_f32_16x16x4_f32(false, x1, false, b,
                                                 (short)0, acc1, false, false);
  }

  float* crow = C + (size_t)(m0 + (lane >> 4) * 8) * DOUT + n0 + mlane;
#pragma unroll
  for (int j = 0; j < 8; ++j)
    crow[(size_t)j * DOUT] = acc0[j];
  if (rows > 16) {
    float* crow1 = crow + (size_t)16 * DOUT;
#pragma unroll
    for (int j = 0; j < 8; ++j)
      crow1[(size_t)j * DOUT] = acc1[j];
  }
}

// ---------------------------------------------------------------------------
// Degree / norm precompute: deg starts at 1 (self loop), +1 per incoming edge,
// then dinv = deg > 0 ? 1/sqrt(deg) : 0.
// ---------------------------------------------------------------------------
__global__ void deg_init(float* __restrict__ deg) {
  int i = blockIdx.x * blockDim.x + threadIdx.x;
  if (i < N_NODES) deg[i] = 1.0f;
}

__global__ void deg_accum(const int* __restrict__ dst, float* __restrict__ deg) {
  int i = blockIdx.x * blockDim.x + threadIdx.x;
  if (i < N_EDGES) atomicAdd(&deg[dst[i]], 1.0f);
}

__global__ void deg_to_dinv(float* __restrict__ deg) {
  int i = blockIdx.x * blockDim.x + threadIdx.x;
  if (i < N_NODES) {
    float d = deg[i];
    deg[i] = d > 0.0f ? (1.0f / sqrtf(d)) : 0.0f;
  }
}

__global__ void zero_buf(float* __restrict__ p, int n) {
  int i = blockIdx.x * blockDim.x + threadIdx.x;
  if (i < n) p[i] = 0.0f;
}

// ---------------------------------------------------------------------------
// Edge scatter: one wave32 per edge; lane handles a float4 of the 128-dim
// feature. Gather is a coalesced 512B read (L2-resident: h is 25.6MB << 192MB
// L2); scatter is 4 global_atomic_add_f32 per lane.
// ---------------------------------------------------------------------------
__launch_bounds__(256)
__global__ void edge_scatter(const float* __restrict__ H,
                             const int* __restrict__ src,
                             const int* __restrict__ dst,
                             const float* __restrict__ dinv,
                             float* __restrict__ agg, int nedges) {
  int e    = (int)((blockIdx.x * 256u + threadIdx.x) >> 5);
  int lane = threadIdx.x & 31;
  if (e >= nedges) return;
  int s = src[e];
  int d = dst[e];
  float nrm = dinv[s] * dinv[d];
  const float4 v = ((const float4*)(H + (size_t)s * DOUT))[lane];
  float* out = agg + (size_t)d * DOUT + (size_t)lane * 4;
  atomicAdd(out + 0, v.x * nrm);
  atomicAdd(out + 1, v.y * nrm);
  atomicAdd(out + 2, v.z * nrm);
  atomicAdd(out + 3, v.w * nrm);
}

// ---------------------------------------------------------------------------
// out = relu(agg + hlin * dinv^2 (self-loop term) + b)
// ---------------------------------------------------------------------------
__global__ void bias_relu(const float* __restrict__ agg,
                          const float* __restrict__ hlin,
                          const float* __restrict__ dinv,
                          const float* __restrict__ b,
                          float* __restrict__ out) {
  int i = blockIdx.x * blockDim.x + threadIdx.x;
  if (i < N_NODES * DOUT) {
    int node = i >> 7;
    int c    = i & (DOUT - 1);
    float di = dinv[node];
    float v  = agg[i] + hlin[i] * di * di + b[c];
    out[i]   = v > 0.0f ? v : 0.0f;
  }
}

extern "C" void kernel_launch(void* const* d_in, const int* in_sizes, int n_in,
                              void* d_out, int out_size, void* d_ws, size_t ws_size,
                              hipStream_t stream) {
  (void)in_sizes; (void)n_in; (void)out_size; (void)ws_size;

  const float* x  = (const float*)d_in[0];
  const int*   ei = (const int*)d_in[1];       // [2, E]: row0 = src, row1 = dst
  const float* Wl[3] = {(const float*)d_in[2], (const float*)d_in[4], (const float*)d_in[6]};
  const float* bl[3] = {(const float*)d_in[3], (const float*)d_in[5], (const float*)d_in[7]};
  const int* src = ei;
  const int* dst = ei + N_EDGES;

  // Workspace layout (256B aligned slices)
  char*  ws   = (char*)d_ws;
  float* dinv = (float*)ws;                                   // 50000 f32
  size_t off  = ((size_t)N_NODES * 4 + 255) & ~(size_t)255;   // 200192
  float* hlin = (float*)(ws + off);                           // 6.4M f32
  float* agg  = (float*)(ws + off + (size_t)N_NODES * DOUT * 4);

  const int NE = N_NODES * DOUT;  // 6,400,000

  // norm precompute
  deg_init   <<<(N_NODES + 255) / 256, 256, 0, stream>>>(dinv);
  deg_accum  <<<(N_EDGES + 255) / 256, 256, 0, stream>>>(dst, dinv);
  deg_to_dinv<<<(N_NODES + 255) / 256, 256, 0, stream>>>(dinv);

  const float* hin  = x;
  float*       hout = (float*)d_out;
  for (int l = 0; l < 3; ++l) {
    gemm_wmma_f32<<<(N_NODES + 31) / 32, 256, 0, stream>>>(hin, Wl[l], hlin, N_NODES);
    zero_buf     <<<(NE + 255) / 256, 256, 0, stream>>>(agg, NE);
    edge_scatter <<<N_EDGES / 8, 256, 0, stream>>>(hlin, src, dst, dinv, agg, N_EDGES);
    bias_relu    <<<(NE + 255) / 256, 256, 0, stream>>>(agg, hlin, dinv, bl[l], hout);
    hin = hout;   // layer l+1 GEMM reads d_out before it is rewritten
  }
}